// ChunkA_16466904613025
// MI455X (gfx1250) — compile-verified
//
#include <hip/hip_runtime.h>
#include <hip/hip_bf16.h>

// ---------------------------------------------------------------------------
// Single-token 4-layer decoder chunk for MI455X (gfx1250).
// Bandwidth-bound fp32 decode: wave-per-row matvecs (b128 loads, x in LDS),
// flash-decoding attention (8 kv-groups x 32 ctx splits) using
// V_WMMA_F32_16X16X4_F32 for both Q*K^T and P*V.
// Round 2: removed EXEC-masked predicated LDS loads around WMMA (zero-row
// trick + register-preloaded A fragments), uniform-branch POS substitution.
// ---------------------------------------------------------------------------

#define LAYERS  4
#define HID     2048
#define NHEADS  16
#define KVHEADS 8
#define HD      128
#define CTXLEN  4096
#define IMLP    6144
#define NSPLIT  32
#define SPLIT_LEN (CTXLEN / NSPLIT)   // 128

typedef __attribute__((ext_vector_type(2))) float v2f;
typedef __attribute__((ext_vector_type(8))) float v8f;

__device__ __forceinline__ float wave_sum(float v) {
#pragma unroll
  for (int off = 16; off > 0; off >>= 1) v += __shfl_xor(v, off, 32);
  return v;
}

// ---------------------------------------------------------------------------
__global__ void k_embed(float* __restrict__ h, const float* __restrict__ embed,
                        const int* __restrict__ ids) {
  int i = blockIdx.x * blockDim.x + threadIdx.x;
  if (i < HID) h[i] = embed[(size_t)ids[0] * HID + i];
}

// x = h * rsqrt(mean(h*h)+eps) * w     (single block)
__global__ void __launch_bounds__(256) k_rms(float* __restrict__ x,
                                             const float* __restrict__ h,
                                             const float* __restrict__ w, int n) {
  __shared__ float red[256];
  __shared__ float rns;
  int tid = threadIdx.x;
  float s = 0.f;
  for (int i = tid; i < n; i += 256) { float v = h[i]; s += v * v; }
  red[tid] = s; __syncthreads();
  for (int off = 128; off > 0; off >>= 1) {
    if (tid < off) red[tid] += red[tid + off];
    __syncthreads();
  }
  if (tid == 0) rns = rsqrtf(red[0] / (float)n + 1e-6f);
  __syncthreads();
  float rn = rns;
  for (int i = tid; i < n; i += 256) x[i] = h[i] * rn * w[i];
}

// fused Q/K/V matvec: 4096 rows, wave per row, x staged in LDS
__global__ void __launch_bounds__(256) k_qkv(float* __restrict__ q,
                                             float* __restrict__ kc,
                                             float* __restrict__ vc,
                                             const float* __restrict__ Wq,
                                             const float* __restrict__ Wk,
                                             const float* __restrict__ Wv,
                                             const float* __restrict__ x) {
  __shared__ __align__(16) float xs[HID];
  int tid = threadIdx.x;
  for (int i = tid; i < HID; i += 256) xs[i] = x[i];
  __syncthreads();
  int wave = tid >> 5, lane = tid & 31;
  int r = blockIdx.x * 8 + wave;
  const float* wrow;
  float* out;
  if (r < HID)                    { wrow = Wq + (size_t)r * HID;                      out = q  + r; }
  else if (r < HID + KVHEADS*HD)  { wrow = Wk + (size_t)(r - HID) * HID;              out = kc + (r - HID); }
  else                            { wrow = Wv + (size_t)(r - HID - KVHEADS*HD) * HID; out = vc + (r - HID - KVHEADS*HD); }
  const float4* w4 = (const float4*)wrow;
  const float4* x4 = (const float4*)xs;
  float s = 0.f;
#pragma unroll 4
  for (int t = 0; t < HID / 128; t++) {
    float4 a = w4[t * 32 + lane];
    float4 b = x4[t * 32 + lane];
    s += a.x * b.x + a.y * b.y + a.z * b.z + a.w * b.w;
  }
  s = wave_sum(s);
  if (lane == 0) *out = s;
}

// per-head RMS norm + RoPE on q (16 heads) and k (8 heads), in place.
// one wave per head; reads raw values before the wave's write (lockstep-safe).
__global__ void __launch_bounds__(256) k_qknr(float* __restrict__ q,
                                              float* __restrict__ kc,
                                              const float* __restrict__ wq,
                                              const float* __restrict__ wk,
                                              const float* __restrict__ cosc,
                                              const float* __restrict__ sinc,
                                              const int* __restrict__ pos_ids) {
  int pos = pos_ids[0];
  const float* cr = cosc + (size_t)pos * HD;
  const float* sr = sinc + (size_t)pos * HD;
  int wave = threadIdx.x >> 5, lane = threadIdx.x & 31;
  for (int job = wave; job < NHEADS + KVHEADS; job += 8) {
    float* base;
    const float* w;
    if (job < NHEADS) { base = q + job * HD;             w = wq; }
    else              { base = kc + (job - NHEADS) * HD; w = wk; }
    int d0 = lane * 4;
    float4 raw = *(const float4*)(base + d0);
    float ss = raw.x * raw.x + raw.y * raw.y + raw.z * raw.z + raw.w * raw.w;
    ss = wave_sum(ss);
    float rn = rsqrtf(ss / (float)HD + 1e-6f);
    int pd = (d0 < 64) ? d0 + 64 : d0 - 64;
    float sgn = (d0 < 64) ? -1.f : 1.f;
    float4 praw = *(const float4*)(base + pd);
    float4 o;
    o.x = raw.x * rn * w[d0 + 0] * cr[d0 + 0] + sgn * praw.x * rn * w[pd + 0] * sr[d0 + 0];
    o.y = raw.y * rn * w[d0 + 1] * cr[d0 + 1] + sgn * praw.y * rn * w[pd + 1] * sr[d0 + 1];
    o.z = raw.z * rn * w[d0 + 2] * cr[d0 + 2] + sgn * praw.z * rn * w[pd + 2] * sr[d0 + 2];
    o.w = raw.w * rn * w[d0 + 3] * cr[d0 + 3] + sgn * praw.w * rn * w[pd + 3] * sr[d0 + 3];
    *(float4*)(base + d0) = o;
  }
}

// ---------------------------------------------------------------------------
// Flash-decoding attention split kernel. Grid = (KVHEADS, NSPLIT), block 256.
// Per block: kv group g, ctx window [sp*128, sp*128+128).
// Q*K^T and P*V via V_WMMA_F32_16X16X4_F32:
//   A (16x4 f32, 2 VGPR): lanes 0-15 -> rows, K=0,1; lanes 16-31 -> K=2,3.
//   B (4x16 f32, 2 VGPR): lanes 0-15 -> K rows 0,1; lanes 16-31 -> K rows 2,3.
//   C/D (16x16 f32, 8 VGPR): VGPR r = row r (lanes 0-15), row r+8 (lanes 16-31).
// A carries the group's 2 q heads zero-padded to M=16 via a zeroed LDS row
// (no EXEC-masked loads); A fragments are preloaded into registers so the
// inner loop is just global-B load + WMMA. The reference's masked KV-cache
// write is emulated by substituting kcur/vcur at c == POS (uniform branch:
// only the split containing POS pays for the per-lane select).
// ---------------------------------------------------------------------------
__global__ void __launch_bounds__(256) k_attn_split(
    float* __restrict__ pmax, float* __restrict__ psum, float* __restrict__ pao,
    const float* __restrict__ q, const float* __restrict__ kcur,
    const float* __restrict__ vcur, const float* __restrict__ Kc,
    const float* __restrict__ Vc, const float* __restrict__ mask,
    const int* __restrict__ pos_ids) {
  __shared__ __align__(16) float qsh[3][HD];        // row 2 = zeros (M-padding)
  __shared__ __align__(16) float sc[3][SPLIT_LEN];  // row 2 = zeros (M-padding)
  __shared__ float red[256];
  __shared__ float smax[2], ssum[2];

  const int g = blockIdx.x;
  const int sp = blockIdx.y;
  const int c_base = sp * SPLIT_LEN;
  const int pos = pos_ids[0];
  const int tid = threadIdx.x, wave = tid >> 5, lane = tid & 31;
  const float scale = 0.08838834764831845f;  // 1/sqrt(128)
  const bool inrange = (pos >= c_base) && (pos < c_base + SPLIT_LEN);

  if (tid < 2 * HD) qsh[tid >> 7][tid & 127] = q[(2 * g) * HD + tid];
  if (tid < HD) qsh[2][tid] = 0.f;
  if (tid < SPLIT_LEN) sc[2][tid] = 0.f;
  __syncthreads();

  const int ln = lane & 15;              // A row m AND B column n
  const int koff = (lane < 16) ? 0 : 2;
  const int arow = (ln < 2) ? ln : 2;    // zero-row padding for m >= 2

  // --- scores: wave w owns ctx tile w (8 tiles of 16 = 128 positions) ------
  {
    const int c0 = c_base + wave * 16;
    // preload A fragments (q) into registers: unconditional ds_load_b64
    v2f afr[HD / 4];
#pragma unroll
    for (int j = 0; j < HD / 4; j++)
      afr[j] = *(const v2f*)&qsh[arow][4 * j + koff];

    const float* krow = Kc + ((size_t)g * CTXLEN + c0 + ln) * HD;
    if (inrange && (c0 + ln == pos)) krow = kcur + g * HD;  // select once

    v8f acc = {0.f, 0.f, 0.f, 0.f, 0.f, 0.f, 0.f, 0.f};
#pragma unroll
    for (int j = 0; j < HD / 4; j++) {
      v2f b = *(const v2f*)(krow + 4 * j + koff);   // single global_load_b64
      acc = __builtin_amdgcn_wmma_f32_16x16x4_f32(false, afr[j], false, b,
                                                  (short)0, acc, false, false);
    }
    if (lane < 16) {
      float mk = mask[c0 + lane];
      sc[0][wave * 16 + lane] = acc[0] * scale + mk;
      sc[1][wave * 16 + lane] = acc[1] * scale + mk;
    }
  }
  __syncthreads();

  // --- local softmax stats (max, sum of exp) over this split ---------------
  for (int h2 = 0; h2 < 2; h2++) {
    float mx = -1e30f;
    for (int c = tid; c < SPLIT_LEN; c += 256) mx = fmaxf(mx, sc[h2][c]);
    red[tid] = mx; __syncthreads();
    for (int off = 128; off > 0; off >>= 1) {
      if (tid < off) red[tid] = fmaxf(red[tid], red[tid + off]);
      __syncthreads();
    }
    if (tid == 0) smax[h2] = red[0];
    __syncthreads();
    float lm = smax[h2];
    float s = 0.f;
    for (int c = tid; c < SPLIT_LEN; c += 256) {
      float e = __expf(sc[h2][c] - lm);
      sc[h2][c] = e;
      s += e;
    }
    red[tid] = s; __syncthreads();
    for (int off = 128; off > 0; off >>= 1) {
      if (tid < off) red[tid] += red[tid + off];
      __syncthreads();
    }
    if (tid == 0) ssum[h2] = red[0];
    __syncthreads();
  }

  // --- partial P*V: wave w owns head-dim tile w (8 tiles of 16 = 128) ------
  {
    const int dt = wave;
    // preload probability A fragments into registers (zero row for m >= 2)
    v2f pfr[SPLIT_LEN / 4];
#pragma unroll
    for (int j = 0; j < SPLIT_LEN / 4; j++)
      pfr[j] = *(const v2f*)&sc[arow][4 * j + koff];

    const float* vbase = Vc + (size_t)g * CTXLEN * HD;
    v8f acc = {0.f, 0.f, 0.f, 0.f, 0.f, 0.f, 0.f, 0.f};
    if (!inrange) {
      // clean path: no POS substitution anywhere in this split
#pragma unroll
      for (int j = 0; j < SPLIT_LEN / 4; j++) {
        int c1 = c_base + 4 * j + koff;
        v2f b;
        b.x = vbase[(size_t)c1 * HD + dt * 16 + ln];
        b.y = vbase[(size_t)(c1 + 1) * HD + dt * 16 + ln];
        acc = __builtin_amdgcn_wmma_f32_16x16x4_f32(false, pfr[j], false, b,
                                                    (short)0, acc, false, false);
      }
    } else {
#pragma unroll
      for (int j = 0; j < SPLIT_LEN / 4; j++) {
        int c1 = c_base + 4 * j + koff;
        int c2 = c1 + 1;
        const float* vr1 = (c1 == pos) ? (vcur + g * HD) : (vbase + (size_t)c1 * HD);
        const float* vr2 = (c2 == pos) ? (vcur + g * HD) : (vbase + (size_t)c2 * HD);
        v2f b;
        b.x = vr1[dt * 16 + ln];
        b.y = vr2[dt * 16 + ln];
        acc = __builtin_amdgcn_wmma_f32_16x16x4_f32(false, pfr[j], false, b,
                                                    (short)0, acc, false, false);
      }
    }
    if (lane < 16) {
      size_t base = ((size_t)g * NSPLIT + sp) * 2;
      pao[(base + 0) * HD + dt * 16 + lane] = acc[0];
      pao[(base + 1) * HD + dt * 16 + lane] = acc[1];
    }
  }
  if (tid < 2) {
    size_t base = ((size_t)g * NSPLIT + sp) * 2;
    pmax[base + tid] = smax[tid];
    psum[base + tid] = ssum[tid];
  }
}

// combine the 32 ctx-split partials per head (standard flash-decode merge)
__global__ void __launch_bounds__(256) k_attn_combine(
    float* __restrict__ ao, const float* __restrict__ pmax,
    const float* __restrict__ psum, const float* __restrict__ pao) {
  int idx = blockIdx.x * 256 + threadIdx.x;   // 0..2047 = head*128 + d
  int hd = idx & (HD - 1);
  int head = idx >> 7;
  int g = head >> 1, hh = head & 1;
  float gmax = -1e30f;
  for (int s = 0; s < NSPLIT; s++)
    gmax = fmaxf(gmax, pmax[((size_t)g * NSPLIT + s) * 2 + hh]);
  float tot = 0.f, acc = 0.f;
  for (int s = 0; s < NSPLIT; s++) {
    size_t b = ((size_t)g * NSPLIT + s) * 2 + hh;
    float f = __expf(pmax[b] - gmax);
    tot += f * psum[b];
    acc += f * pao[b * HD + hd];
  }
  ao[idx] = acc / tot;
}

// h[r] += dot(W[r,:], x)   (wave per row, x staged in LDS)
template <int COLS>
__global__ void __launch_bounds__(256) k_matvec_add(float* __restrict__ h,
                                                    const float* __restrict__ W,
                                                    const float* __restrict__ x) {
  __shared__ __align__(16) float xs[COLS];
  int tid = threadIdx.x;
  for (int i = tid; i < COLS; i += 256) xs[i] = x[i];
  __syncthreads();
  int wave = tid >> 5, lane = tid & 31;
  int r = blockIdx.x * 8 + wave;
  const float4* w4 = (const float4*)(W + (size_t)r * COLS);
  const float4* x4 = (const float4*)xs;
  float s = 0.f;
#pragma unroll 4
  for (int t = 0; t < COLS / 128; t++) {
    float4 a = w4[t * 32 + lane];
    float4 b = x4[t * 32 + lane];
    s += a.x * b.x + a.y * b.y + a.z * b.z + a.w * b.w;
  }
  s = wave_sum(s);
  if (lane == 0) h[r] += s;
}

// t[r] = silu(Wg[r]·x) * (Wu[r]·x)   (wave per row, both dots fused)
__global__ void __launch_bounds__(256) k_mlp_gu(float* __restrict__ t,
                                                const float* __restrict__ Wg,
                                                const float* __restrict__ Wu,
                                                const float* __restrict__ x) {
  __shared__ __align__(16) float xs[HID];
  int tid = threadIdx.x;
  for (int i = tid; i < HID; i += 256) xs[i] = x[i];
  __syncthreads();
  int wave = tid >> 5, lane = tid & 31;
  int r = blockIdx.x * 8 + wave;
  const float4* g4 = (const float4*)(Wg + (size_t)r * HID);
  const float4* u4 = (const float4*)(Wu + (size_t)r * HID);
  const float4* x4 = (const float4*)xs;
  float sg = 0.f, su = 0.f;
#pragma unroll 2
  for (int tt = 0; tt < HID / 128; tt++) {
    float4 b = x4[tt * 32 + lane];
    float4 a = g4[tt * 32 + lane];
    float4 c = u4[tt * 32 + lane];
    sg += a.x * b.x + a.y * b.y + a.z * b.z + a.w * b.w;
    su += c.x * b.x + c.y * b.y + c.z * b.z + c.w * b.w;
  }
  sg = wave_sum(sg);
  su = wave_sum(su);
  if (lane == 0) t[r] = (sg / (1.f + __expf(-sg))) * su;
}

__global__ void k_copy(float* __restrict__ out, const float* __restrict__ h) {
  int i = blockIdx.x * 256 + threadIdx.x;
  if (i < HID) out[i] = h[i];
}

// ---------------------------------------------------------------------------
extern "C" void kernel_launch(void* const* d_in, const int* in_sizes, int n_in,
                              void* d_out, int out_size, void* d_ws, size_t ws_size,
                              hipStream_t stream) {
  (void)in_sizes; (void)n_in; (void)out_size; (void)ws_size;
  const int*   ids   = (const int*)d_in[0];
  const int*   pos   = (const int*)d_in[1];
  const float* mask  = (const float*)d_in[2];
  // d_in[3] = update_mask: unused (POS read directly from position_ids)
  const float* embed = (const float*)d_in[4];
  const float* cosc  = (const float*)d_in[5];
  const float* sinc  = (const float*)d_in[6];
  const float* kvc   = (const float*)d_in[7];
  const float* wln1  = (const float*)d_in[8];
  const float* wln2  = (const float*)d_in[9];
  const float* wqn   = (const float*)d_in[10];
  const float* wkn   = (const float*)d_in[11];
  const float* Wq    = (const float*)d_in[12];
  const float* Wk    = (const float*)d_in[13];
  const float* Wv    = (const float*)d_in[14];
  const float* Wo    = (const float*)d_in[15];
  const float* Wg    = (const float*)d_in[16];
  const float* Wu    = (const float*)d_in[17];
  const float* Wd    = (const float*)d_in[18];

  float* ws   = (float*)d_ws;
  float* h    = ws;                 // [2048]
  float* x    = ws + 2048;          // [2048]
  float* q    = ws + 4096;          // [2048]
  float* kcur = ws + 6144;          // [1024]
  float* vcur = ws + 7168;          // [1024]
  float* ao   = ws + 8192;          // [2048]
  float* t    = ws + 10240;         // [6144]
  float* pmax = ws + 16384;         // [512]
  float* psum = ws + 16896;         // [512]
  float* pao  = ws + 17408;         // [65536]

  k_embed<<<8, 256, 0, stream>>>(h, embed, ids);
  for (int i = 0; i < LAYERS; i++) {
    const float* Kc = kvc + (size_t)i * KVHEADS * CTXLEN * HD;
    const float* Vc = kvc + (size_t)(LAYERS + i) * KVHEADS * CTXLEN * HD;
    k_rms<<<1, 256, 0, stream>>>(x, h, wln1 + i * HID, HID);
    k_qkv<<<512, 256, 0, stream>>>(q, kcur, vcur,
                                   Wq + (size_t)i * HID * HID,
                                   Wk + (size_t)i * KVHEADS * HD * HID,
                                   Wv + (size_t)i * KVHEADS * HD * HID, x);
    k_qknr<<<1, 256, 0, stream>>>(q, kcur, wqn + i * HD, wkn + i * HD,
                                  cosc, sinc, pos);
    dim3 agrid(KVHEADS, NSPLIT);
    k_attn_split<<<agrid, 256, 0, stream>>>(pmax, psum, pao, q, kcur, vcur,
                                            Kc, Vc, mask, pos);
    k_attn_combine<<<8, 256, 0, stream>>>(ao, pmax, psum, pao);
    k_matvec_add<2048><<<256, 256, 0, stream>>>(h, Wo + (size_t)i * HID * HID, ao);
    k_rms<<<1, 256, 0, stream>>>(x, h, wln2 + i * HID, HID);
    k_mlp_gu<<<768, 256, 0, stream>>>(t, Wg + (size_t)i * IMLP * HID,
                                      Wu + (size_t)i * IMLP * HID, x);
    k_matvec_add<6144><<<256, 256, 0, stream>>>(h, Wd + (size_t)i * HID * IMLP, t);
  }
  k_copy<<<8, 256, 0, stream>>>((float*)d_out, h);
}